// VoxelTripletLoss_39152921870432
// MI455X (gfx1250) — compile-verified
//
#include <hip/hip_runtime.h>
#include <hip/hip_bf16.h>
#include <math.h>

typedef __attribute__((ext_vector_type(2))) float v2f;
typedef __attribute__((ext_vector_type(8))) float v8f;

#define BROWS 64
#define MMEM  2048
#define FEAT  512
#define NVOX  32768
#define KCHUNKS 8
#define KCHUNK (NVOX / KCHUNKS)

#define MARGIN 0.2f
#define BETA   0.8333333f
#define EPS_N  1e-8f
#define BIGV   1.0e9f

// ---------------------------------------------------------------------------
// Row sum-of-squares (mode 0) or clamped L2 norm (mode 1). One block per row.
// ---------------------------------------------------------------------------
__global__ __launch_bounds__(256)
void rowsumsq_kernel(const float* __restrict__ src, float* __restrict__ dst,
                     int cols, int mode) {
    const int row = blockIdx.x;
    const int tid = threadIdx.x;
    const float* p = src + (size_t)row * cols;
    float s = 0.f;
    for (int c = tid; c < cols; c += 256) {
        float x = p[c];
        s = fmaf(x, x, s);
    }
    __shared__ float red[256];
    red[tid] = s;
    __syncthreads();
    for (int off = 128; off > 0; off >>= 1) {
        if (tid < off) red[tid] += red[tid + off];
        __syncthreads();
    }
    if (tid == 0) {
        float v = red[0];
        dst[row] = mode ? fmaxf(sqrtf(v), EPS_N) : v;
    }
}

// ---------------------------------------------------------------------------
// Voxel GEMM: Gpart[kc] += v(64xN) * w(2048xN)^T over one K chunk, fp32 WMMA.
// Wave = one 16-wide column tile (16 rows of w), all 4 row tiles of v.
// w_sq partial sums fused into the streaming pass over w.
// Grid: (32, KCHUNKS) blocks x 128 threads (4 waves/block).
// ---------------------------------------------------------------------------
__global__ __launch_bounds__(128)
void voxel_gemm_kernel(const float* __restrict__ v, const float* __restrict__ w,
                       float* __restrict__ Gpart, float* __restrict__ Wsqpart) {
    const int lane  = threadIdx.x & 31;
    const int wave  = threadIdx.x >> 5;
    const int ctile = blockIdx.x * 4 + wave;       // 0..127
    const int m0    = ctile * 16;
    const int kc    = blockIdx.y;                  // 0..7
    const int kbeg  = kc * KCHUNK;
    const int kend  = kbeg + KCHUNK;

    const int r    = lane & 15;                    // row within tile
    const int koff = (lane >> 4) << 1;             // 0 or 2 (K sub-offset)

    const float* wrow  = w + (size_t)(m0 + r) * NVOX + koff;   // B fragment src
    const float* vrow  = v + (size_t)r * NVOX + koff;          // A fragment src

    v8f acc0 = {}, acc1 = {}, acc2 = {}, acc3 = {};
    float wsq = 0.f;

    for (int k = kbeg; k < kend; k += 32) {
        // stream w ~4KB ahead (emits global_prefetch_b8)
        __builtin_prefetch(wrow + k + 1024, 0, 1);
#pragma unroll
        for (int kk = 0; kk < 32; kk += 4) {
            v2f bf = *(const v2f*)(wrow + k + kk);
            wsq = fmaf(bf.x, bf.x, wsq);
            wsq = fmaf(bf.y, bf.y, wsq);
            v2f a0 = *(const v2f*)(vrow + k + kk);
            v2f a1 = *(const v2f*)(vrow + 16 * NVOX + k + kk);
            v2f a2 = *(const v2f*)(vrow + 32 * NVOX + k + kk);
            v2f a3 = *(const v2f*)(vrow + 48 * NVOX + k + kk);
            acc0 = __builtin_amdgcn_wmma_f32_16x16x4_f32(false, a0, false, bf,
                                                         (short)0, acc0, false, false);
            acc1 = __builtin_amdgcn_wmma_f32_16x16x4_f32(false, a1, false, bf,
                                                         (short)0, acc1, false, false);
            acc2 = __builtin_amdgcn_wmma_f32_16x16x4_f32(false, a2, false, bf,
                                                         (short)0, acc2, false, false);
            acc3 = __builtin_amdgcn_wmma_f32_16x16x4_f32(false, a3, false, bf,
                                                         (short)0, acc3, false, false);
        }
    }

    // Store 16x16 tiles: VGPR g of C/D holds row (g + 8*(lane>>4)), col lane&15.
    const int hi = lane >> 4;
    const int lo = lane & 15;
    float* gout = Gpart + (size_t)kc * BROWS * MMEM;
#pragma unroll
    for (int g = 0; g < 8; ++g) {
        const int row = g + 8 * hi;
        gout[(size_t)(row     ) * MMEM + m0 + lo] = acc0[g];
        gout[(size_t)(row + 16) * MMEM + m0 + lo] = acc1[g];
        gout[(size_t)(row + 32) * MMEM + m0 + lo] = acc2[g];
        gout[(size_t)(row + 48) * MMEM + m0 + lo] = acc3[g];
    }

    // Fused w_sq: lanes r and r+16 each hold half of row m0+r's K values.
    float other = __shfl_xor(wsq, 16, 32);
    if (lane < 16)
        Wsqpart[kc * MMEM + m0 + lane] = wsq + other;
}

// ---------------------------------------------------------------------------
// Cosine GEMM: key_dists = 1 - (anchor . keys) / (||a|| * ||k||). K=512.
// Grid: 32 blocks x 128 threads (4 waves -> 128 column tiles).
// ---------------------------------------------------------------------------
__global__ __launch_bounds__(128)
void cos_gemm_kernel(const float* __restrict__ a, const float* __restrict__ keys,
                     const float* __restrict__ an, const float* __restrict__ kn,
                     float* __restrict__ kd) {
    const int lane  = threadIdx.x & 31;
    const int wave  = threadIdx.x >> 5;
    const int ctile = blockIdx.x * 4 + wave;
    const int m0    = ctile * 16;

    const int r    = lane & 15;
    const int koff = (lane >> 4) << 1;

    const float* krow = keys + (size_t)(m0 + r) * FEAT + koff;
    const float* arow = a + (size_t)r * FEAT + koff;

    v8f acc0 = {}, acc1 = {}, acc2 = {}, acc3 = {};
#pragma unroll 8
    for (int k = 0; k < FEAT; k += 4) {
        v2f bf = *(const v2f*)(krow + k);
        v2f a0 = *(const v2f*)(arow + k);
        v2f a1 = *(const v2f*)(arow + 16 * FEAT + k);
        v2f a2 = *(const v2f*)(arow + 32 * FEAT + k);
        v2f a3 = *(const v2f*)(arow + 48 * FEAT + k);
        acc0 = __builtin_amdgcn_wmma_f32_16x16x4_f32(false, a0, false, bf,
                                                     (short)0, acc0, false, false);
        acc1 = __builtin_amdgcn_wmma_f32_16x16x4_f32(false, a1, false, bf,
                                                     (short)0, acc1, false, false);
        acc2 = __builtin_amdgcn_wmma_f32_16x16x4_f32(false, a2, false, bf,
                                                     (short)0, acc2, false, false);
        acc3 = __builtin_amdgcn_wmma_f32_16x16x4_f32(false, a3, false, bf,
                                                     (short)0, acc3, false, false);
    }

    const int hi = lane >> 4;
    const int lo = lane & 15;
    const float knv = kn[m0 + lo];
#pragma unroll
    for (int g = 0; g < 8; ++g) {
        const int row = g + 8 * hi;
        kd[(size_t)(row     ) * MMEM + m0 + lo] = 1.f - acc0[g] / (an[row     ] * knv);
        kd[(size_t)(row + 16) * MMEM + m0 + lo] = 1.f - acc1[g] / (an[row + 16] * knv);
        kd[(size_t)(row + 32) * MMEM + m0 + lo] = 1.f - acc2[g] / (an[row + 32] * knv);
        kd[(size_t)(row + 48) * MMEM + m0 + lo] = 1.f - acc3[g] / (an[row + 48] * knv);
    }
}

// ---------------------------------------------------------------------------
// Per-anchor hardest-pos / hardest-neg mining. One block per anchor row.
// Sums the 8 K-chunk partials deterministically here.
// ---------------------------------------------------------------------------
__global__ __launch_bounds__(256)
void rowloss_kernel(const float* __restrict__ Gpart,
                    const float* __restrict__ Wsqpart,
                    const float* __restrict__ vsq,
                    const float* __restrict__ kd,
                    float* __restrict__ lossArr, float* __restrict__ validArr) {
    const int b   = blockIdx.x;
    const int tid = threadIdx.x;
    const float vb = vsq[b];
    const float invN = 1.0f / (float)NVOX;

    float hp = -BIGV, hn = BIGV;
    float anyp = 0.f, anyn = 0.f;
    for (int m = tid; m < MMEM; m += 256) {
        float g = 0.f, wsq = 0.f;
#pragma unroll
        for (int c = 0; c < KCHUNKS; ++c) {
            g   += Gpart[(size_t)c * BROWS * MMEM + (size_t)b * MMEM + m];
            wsq += Wsqpart[c * MMEM + m];
        }
        const float mse = (vb + wsq - 2.f * g) * invN;
        const float sv  = 1.f - mse;
        const float d   = kd[(size_t)b * MMEM + m];
        if (sv > BETA) { anyp = 1.f; hp = fmaxf(hp, d); }
        if (sv < BETA) { anyn = 1.f; hn = fminf(hn, d); }
    }

    __shared__ float shp[256], shn[256], sap[256], san[256];
    shp[tid] = hp; shn[tid] = hn; sap[tid] = anyp; san[tid] = anyn;
    __syncthreads();
    for (int off = 128; off > 0; off >>= 1) {
        if (tid < off) {
            shp[tid] = fmaxf(shp[tid], shp[tid + off]);
            shn[tid] = fminf(shn[tid], shn[tid + off]);
            sap[tid] = fmaxf(sap[tid], sap[tid + off]);
            san[tid] = fmaxf(san[tid], san[tid + off]);
        }
        __syncthreads();
    }
    if (tid == 0) {
        const float loss  = fmaxf(shp[0] - shn[0] + MARGIN, 0.f);
        const float valid = (sap[0] > 0.f && san[0] > 0.f) ? 1.f : 0.f;
        lossArr[b]  = valid * loss;
        validArr[b] = valid;
    }
}

__global__ __launch_bounds__(64)
void finalize_kernel(const float* __restrict__ lossArr,
                     const float* __restrict__ validArr,
                     float* __restrict__ out) {
    __shared__ float st[64], sc[64];
    const int t = threadIdx.x;
    st[t] = lossArr[t];
    sc[t] = validArr[t];
    __syncthreads();
    for (int off = 32; off > 0; off >>= 1) {
        if (t < off) { st[t] += st[t + off]; sc[t] += sc[t + off]; }
        __syncthreads();
    }
    if (t == 0)
        out[0] = (sc[0] > 0.f) ? (st[0] / fmaxf(sc[0], 1.f)) : 0.f;
}

// ---------------------------------------------------------------------------
extern "C" void kernel_launch(void* const* d_in, const int* in_sizes, int n_in,
                              void* d_out, int out_size, void* d_ws, size_t ws_size,
                              hipStream_t stream) {
    (void)in_sizes; (void)n_in; (void)out_size; (void)ws_size;
    const float* anchor  = (const float*)d_in[0];  // (64, 512)
    const float* voxels  = (const float*)d_in[1];  // (64, 32, 32, 32)
    const float* keys    = (const float*)d_in[2];  // (2048, 512)
    const float* memvals = (const float*)d_in[3];  // (2048, 32, 32, 32)
    float* out = (float*)d_out;

    float* ws       = (float*)d_ws;
    float* Gpart    = ws;                                   // 8*64*2048
    float* Wsqpart  = Gpart   + (size_t)KCHUNKS * BROWS * MMEM;  // 8*2048
    float* kd       = Wsqpart + (size_t)KCHUNKS * MMEM;          // 64*2048
    float* vsq      = kd      + (size_t)BROWS * MMEM;            // 64
    float* an       = vsq     + BROWS;                           // 64
    float* kn       = an      + BROWS;                           // 2048
    float* lossArr  = kn      + MMEM;                            // 64
    float* validArr = lossArr + BROWS;                           // 64

    rowsumsq_kernel<<<BROWS, 256, 0, stream>>>(voxels, vsq, NVOX, 0);
    rowsumsq_kernel<<<BROWS, 256, 0, stream>>>(anchor, an, FEAT, 1);
    rowsumsq_kernel<<<MMEM, 256, 0, stream>>>(keys, kn, FEAT, 1);

    voxel_gemm_kernel<<<dim3(32, KCHUNKS), 128, 0, stream>>>(voxels, memvals,
                                                             Gpart, Wsqpart);
    cos_gemm_kernel<<<dim3(32, 1), 128, 0, stream>>>(anchor, keys, an, kn, kd);

    rowloss_kernel<<<BROWS, 256, 0, stream>>>(Gpart, Wsqpart, vsq, kd,
                                              lossArr, validArr);
    finalize_kernel<<<1, 64, 0, stream>>>(lossArr, validArr, out);
}